// Net_56118042689681
// MI455X (gfx1250) — compile-verified
//
#include <hip/hip_runtime.h>
#include <math.h>

typedef __attribute__((ext_vector_type(2))) float v2f;
typedef __attribute__((ext_vector_type(8))) float v8f;
typedef __attribute__((ext_vector_type(4))) int   v4i;

#define GLOBAL_AS __attribute__((address_space(1)))
#define LDS_AS    __attribute__((address_space(3)))

#define N_NODES 100000
#define F_IN    256
#define H_DIM   128
#define C_DIM   16
#define XPITCH  260   // padded LDS row stride (floats): spreads lanes over all 64 banks

#if __has_builtin(__builtin_amdgcn_global_load_async_to_lds_b128)
#define USE_ASYNC_LDS 1
#endif

__device__ __forceinline__ void wait_staging() {
#ifdef USE_ASYNC_LDS
#if __has_builtin(__builtin_amdgcn_s_wait_asynccnt)
  __builtin_amdgcn_s_wait_asynccnt(0);
#else
  asm volatile("s_wait_asynccnt 0x0" ::: "memory");
#endif
#endif
}

// Stage one 16x256 f32 x-tile (contiguous 16KB in HBM) into a padded LDS buffer.
// Async path: GLOBAL_LOAD_ASYNC_TO_LDS_B128 (per-lane 16B, no VGPR round-trip).
__device__ __forceinline__ void stage_tile(const float* __restrict__ x, int t,
                                           float* sbuf, int tid) {
  const float4* src = (const float4*)(x + (size_t)t * (16 * F_IN));
#pragma unroll
  for (int k = 0; k < 4; ++k) {
    int i    = tid + 256 * k;
    int row  = i >> 6;   // /(F_IN/4)
    int col4 = i & 63;
    float* dst = sbuf + row * XPITCH + col4 * 4;
#ifdef USE_ASYNC_LDS
    __builtin_amdgcn_global_load_async_to_lds_b128(
        (GLOBAL_AS v4i*)(src + i), (LDS_AS v4i*)dst, 0, 0);
#else
    *(float4*)dst = src[i];
#endif
  }
}

// ---------- kernel 1: degree -> D^-1/2 ----------
__global__ void k_dinv(const int* __restrict__ rowptr, const float* __restrict__ ew,
                       float* __restrict__ dinv, int n) {
  int r = blockIdx.x * blockDim.x + threadIdx.x;
  if (r >= n) return;
  int s = rowptr[r], e = rowptr[r + 1];
  float deg = 0.f;
  for (int i = s; i < e; ++i) deg += ew[i];
  dinv[r] = (deg > 0.f) ? rsqrtf(deg) : 0.f;
}

// ---------- kernel 2: h1 = x @ W1 via V_WMMA_F32_16X16X4_F32 ----------
// Block = 256 threads = 8 waves; wave w owns output cols [16w,16w+16).
// W1 fragments in VGPRs; x tiles double-buffered in LDS. The prefetch for the
// NEXT tile is issued after the barrier, so the conservative pre-barrier wait
// at the next iteration only drains a transfer that overlapped 64 WMMAs.
__global__ void __launch_bounds__(256, 1)
k_gemm1(const float* __restrict__ x, const float* __restrict__ W1,
        float* __restrict__ h1, int n_tiles) {
  __shared__ float sX[2][16 * XPITCH];  // ~33 KB

  const int tid  = threadIdx.x;
  const int lane = tid & 31;
  const int wv   = tid >> 5;    // N-tile id (0..7)
  const int half = lane >> 4;   // K-pair select
  const int mr   = lane & 15;   // M row (A) / N col (B)

  // Preload B fragments from W1[256][128]:
  // b[kk].x = W1[4kk+2*half][16wv+mr], b[kk].y = W1[4kk+2*half+1][16wv+mr]
  v2f b[64];
#pragma unroll
  for (int kk = 0; kk < 64; ++kk) {
    const float* p = W1 + (4 * kk + 2 * half) * H_DIM + 16 * wv + mr;
    v2f t; t.x = p[0]; t.y = p[H_DIM];
    b[kk] = t;
  }

  const int t0 = blockIdx.x, stride = gridDim.x;
  if (t0 < n_tiles) stage_tile(x, t0, sX[0], tid);
  int cur = 0;

  for (int t = t0; t < n_tiles; t += stride) {
    wait_staging();   // drain staging of sX[cur] (prefetch overlapped compute)
    __syncthreads();  // staged tile visible to all waves

    const int tn = t + stride;
    if (tn < n_tiles)                         // block-uniform
      stage_tile(x, tn, sX[cur ^ 1], tid);    // async: overlaps WMMAs below

    const float* aB = &sX[cur][mr * XPITCH + 2 * half];
    v8f c = {};
    v2f a = *(const v2f*)aB;
#pragma unroll
    for (int kk = 0; kk < 64; ++kk) {
      v2f an = a;
      if (kk < 63) an = *(const v2f*)(aB + 4 * (kk + 1));  // pipeline next frag
      c = __builtin_amdgcn_wmma_f32_16x16x4_f32(false, a, false, b[kk],
                                                (short)0, c, false, false);
      a = an;
    }
    // C/D layout: VGPR v, lane -> [M = v + 8*half][N = mr]
    float* orow = h1 + (size_t)(t * 16 + 8 * half) * H_DIM + 16 * wv + mr;
#pragma unroll
    for (int v = 0; v < 8; ++v) orow[v * H_DIM] = c[v];

    cur ^= 1;  // next iter's wait+barrier orders restage vs. this iter's reads
  }
}

// ---------- kernel 3: z = relu(Ahat @ h1 + b1), one wave per row ----------
__global__ void k_spmm1(const int* __restrict__ rowptr, const int* __restrict__ colind,
                        const float* __restrict__ ew, const float* __restrict__ dinv,
                        const float* __restrict__ h1, const float* __restrict__ b1,
                        float* __restrict__ z, int n) {
  int row = (int)((blockIdx.x * (unsigned)blockDim.x + threadIdx.x) >> 5);
  if (row >= n) return;
  const int lane = threadIdx.x & 31;  // lane owns features [4*lane, 4*lane+4)
  const int s = rowptr[row], e = rowptr[row + 1];
  const float di = dinv[row];
  float a0 = 0.f, a1 = 0.f, a2 = 0.f, a3 = 0.f;
  for (int i = s; i < e; ++i) {
    int c = colind[i];
    float w = di * ew[i] * dinv[c];
    float4 v = ((const float4*)(h1 + (size_t)c * H_DIM))[lane];  // 512B/wave, L2-hot
    a0 += w * v.x; a1 += w * v.y; a2 += w * v.z; a3 += w * v.w;
  }
  float4 bb = ((const float4*)b1)[lane];
  float4 r;
  r.x = fmaxf(a0 + bb.x, 0.f);
  r.y = fmaxf(a1 + bb.y, 0.f);
  r.z = fmaxf(a2 + bb.z, 0.f);
  r.w = fmaxf(a3 + bb.w, 0.f);
  ((float4*)(z + (size_t)row * H_DIM))[lane] = r;
}

// ---------- kernel 4: y = z @ W2 (one wave per 16-row M-tile, N=16) ----------
__global__ void __launch_bounds__(256, 1)
k_gemm2(const float* __restrict__ z, const float* __restrict__ W2,
        float* __restrict__ y, int n_tiles) {
  const int lane = threadIdx.x & 31;
  const int half = lane >> 4;
  const int mr   = lane & 15;
  const int t    = (int)((blockIdx.x * (unsigned)blockDim.x + threadIdx.x) >> 5);

  v2f b[32];
#pragma unroll
  for (int kk = 0; kk < 32; ++kk) {
    const float* p = W2 + (4 * kk + 2 * half) * C_DIM + mr;
    v2f tt; tt.x = p[0]; tt.y = p[C_DIM];
    b[kk] = tt;
  }
  if (t >= n_tiles) return;  // wave-uniform: EXEC stays all-ones for WMMA

  const int m0 = t * 16;
  const float* arow = z + (size_t)(m0 + mr) * H_DIM + 2 * half;
  v8f c = {};
#pragma unroll
  for (int kb = 0; kb < 4; ++kb) {   // batch A loads: one wait per 8 WMMAs
    v2f av[8];
#pragma unroll
    for (int u = 0; u < 8; ++u) av[u] = *(const v2f*)(arow + 4 * (8 * kb + u));
#pragma unroll
    for (int u = 0; u < 8; ++u)
      c = __builtin_amdgcn_wmma_f32_16x16x4_f32(false, av[u], false, b[8 * kb + u],
                                                (short)0, c, false, false);
  }
  float* orow = y + (size_t)(m0 + 8 * half) * C_DIM + mr;
#pragma unroll
  for (int v = 0; v < 8; ++v) orow[v * C_DIM] = c[v];
}

// ---------- kernel 5: out = log_softmax(Ahat @ y + b2), thread per row ----------
__global__ void k_spmm2_lsm(const int* __restrict__ rowptr, const int* __restrict__ colind,
                            const float* __restrict__ ew, const float* __restrict__ dinv,
                            const float* __restrict__ y, const float* __restrict__ b2,
                            float* __restrict__ out, int n) {
  int r = blockIdx.x * blockDim.x + threadIdx.x;
  if (r >= n) return;
  float acc[16];
#pragma unroll
  for (int j = 0; j < 16; ++j) acc[j] = 0.f;
  const int s = rowptr[r], e = rowptr[r + 1];
  const float di = dinv[r];
  for (int i = s; i < e; ++i) {
    int c = colind[i];
    float w = di * ew[i] * dinv[c];
    const float4* yp = (const float4*)(y + (size_t)c * C_DIM);  // 64B row
#pragma unroll
    for (int j = 0; j < 4; ++j) {
      float4 v = yp[j];
      acc[4 * j + 0] += w * v.x;
      acc[4 * j + 1] += w * v.y;
      acc[4 * j + 2] += w * v.z;
      acc[4 * j + 3] += w * v.w;
    }
  }
  float m = -INFINITY;
#pragma unroll
  for (int j = 0; j < 16; ++j) { acc[j] += b2[j]; m = fmaxf(m, acc[j]); }
  float sum = 0.f;
#pragma unroll
  for (int j = 0; j < 16; ++j) sum += expf(acc[j] - m);
  const float lse = m + logf(sum);
  float* op = out + (size_t)r * C_DIM;
#pragma unroll
  for (int j = 0; j < 16; ++j) op[j] = acc[j] - lse;
}

extern "C" void kernel_launch(void* const* d_in, const int* in_sizes, int n_in,
                              void* d_out, int out_size, void* d_ws, size_t ws_size,
                              hipStream_t stream) {
  const float* x      = (const float*)d_in[0];
  const float* ew     = (const float*)d_in[1];
  const float* W1     = (const float*)d_in[2];
  const float* b1     = (const float*)d_in[3];
  const float* W2     = (const float*)d_in[4];
  const float* b2     = (const float*)d_in[5];
  const int*   rowptr = (const int*)d_in[6];
  const int*   colind = (const int*)d_in[7];
  float* out = (float*)d_out;
  float* ws  = (float*)d_ws;

  const int n = N_NODES;

  // Workspace layout (floats): dinv[n] | h1[n*128] | z[n*128]; y aliases h1.
  float* dinv = ws;                                  // n floats
  float* h1   = ws + 100096;                         // 64B-aligned offset
  float* z    = h1 + (size_t)N_NODES * H_DIM;
  float* y    = h1;                                  // h1 dead after spmm1

  k_dinv<<<(n + 255) / 256, 256, 0, stream>>>(rowptr, ew, dinv, n);
  k_gemm1<<<1024, 256, 0, stream>>>(x, W1, h1, n / 16);
  k_spmm1<<<(n * 32 + 255) / 256, 256, 0, stream>>>(rowptr, colind, ew, dinv, h1, b1, z, n);
  k_gemm2<<<((n / 16) + 7) / 8, 256, 0, stream>>>(z, W2, y, n / 16);
  k_spmm2_lsm<<<(n + 255) / 256, 256, 0, stream>>>(rowptr, colind, ew, dinv, y, b2, out, n);
}